// MultiTimescaleAttention_12618613915813
// MI455X (gfx1250) — compile-verified
//
#include <hip/hip_runtime.h>
#include <math.h>

// ---- problem constants (match the reference) ----
#define TSC 3          // timescales
#define BB  2          // batch
#define SSQ 1024       // sequence length
#define DDM 512        // d_model
#define HHN 8          // heads
#define HDM 64         // head dim
#define DH2 1024       // 2*D
#define DH3 1536       // 3*D

typedef float v2f __attribute__((ext_vector_type(2)));
typedef float v8f __attribute__((ext_vector_type(8)));

__device__ __forceinline__ v8f wmma_f32_4(v2f a, v2f b, v8f c) {
  // (neg_a, A, neg_b, B, c_mod, C, reuse_a, reuse_b)
  return __builtin_amdgcn_wmma_f32_16x16x4_f32(false, a, false, b, (short)0, c,
                                               false, false);
}

// ---------------------------------------------------------------------------
// fp32 WMMA GEMM: C[M,N] = A[M,K] * B[K,N] + bias[N]
// One wave computes a 16x64 tile of C (4 accumulators share the A fragment).
// N must be a multiple of 64.
// ---------------------------------------------------------------------------
__global__ __launch_bounds__(32) void gemm16x64_f32(
    const float* __restrict__ A, const float* __restrict__ Bm,
    const float* __restrict__ bias, float* __restrict__ C,
    int M, int N, int K) {
  const int tilesN = N >> 6;
  const int tm = blockIdx.x / tilesN;
  const int tn = blockIdx.x - tm * tilesN;
  const int m0 = tm << 4;
  const int n0 = tn << 6;
  const int lane = threadIdx.x;
  const int l  = lane & 15;
  const int hf = lane >> 4;   // 0: K {0,1}, 1: K {2,3} within each 4-chunk

  const float* arow  = A + (size_t)(m0 + l) * K + 2 * hf;
  const float* bbase = Bm + (size_t)(2 * hf) * N + n0 + l;

  v8f acc[4] = {};
#pragma unroll 2
  for (int k = 0; k < K; k += 4) {
    v2f a = *(const v2f*)(arow + k);            // A[m][k+2hf], A[m][k+2hf+1]
    const float* bk_ = bbase + (size_t)k * N;
    v2f b0, b1, b2, b3;
    b0.x = bk_[0];      b0.y = bk_[N];
    b1.x = bk_[16];     b1.y = bk_[N + 16];
    b2.x = bk_[32];     b2.y = bk_[N + 32];
    b3.x = bk_[48];     b3.y = bk_[N + 48];
    acc[0] = wmma_f32_4(a, b0, acc[0]);
    acc[1] = wmma_f32_4(a, b1, acc[1]);
    acc[2] = wmma_f32_4(a, b2, acc[2]);
    acc[3] = wmma_f32_4(a, b3, acc[3]);
  }

#pragma unroll
  for (int nt = 0; nt < 4; ++nt) {
    const int col = n0 + nt * 16 + l;
    const float bv = bias ? bias[col] : 0.0f;
#pragma unroll
    for (int r = 0; r < 8; ++r) {
      const int row = m0 + r + 8 * hf;          // C layout: VGPR r -> M = r + 8*half
      C[(size_t)row * N + col] = acc[nt][r] + bv;
    }
  }
}

// ---------------------------------------------------------------------------
// Multi-timescale attention.
// Grid: (S/16 q-tiles, H, T*B). Block: 128 threads (4 waves).
// Dynamic LDS: 16*1024 energy + 32 reduction + 16*64 Q-tile floats (~70 KB).
// Q tile is staged LDS-resident once via global_load_async_to_lds_b128.
// ---------------------------------------------------------------------------
__global__ __launch_bounds__(128) void attn_kernel(
    const float* __restrict__ Qb, const float* __restrict__ Kb,
    const float* __restrict__ Vb, const int* __restrict__ mask,
    const float* __restrict__ gammas,
    float* __restrict__ attn_out, float* __restrict__ xout) {
  extern __shared__ float smem[];
  float* E   = smem;                 // [16][1024] energy / probabilities
  float* red = smem + 16 * SSQ;      // [32]: per-row max, per-row 1/sum
  float* Qt  = red + 32;             // [16][64] staged Q tile

  const int q0 = blockIdx.x << 4;
  const int h  = blockIdx.y;
  const int tb = blockIdx.z;
  const int t  = tb / BB;
  const int b  = tb - t * BB;

  const int tid  = threadIdx.x;
  const int w    = tid >> 5;
  const int lane = tid & 31;
  const int l    = lane & 15;
  const int hf   = lane >> 4;

  const size_t base = (size_t)(t * BB + b) * SSQ * DDM;  // [T,B,S,D]
  const float* Qp = Qb + base;
  const float* Kp = Kb + base;
  const float* Vp = Vb + base;
  const int hoff = h * HDM;

  const float lg = log2f(gammas[t]);
  const float inv_scale = 0.125f;                // 1/sqrt(64)
  const int* mrow = mask + b * SSQ;

  // ---- stage Q tile [16][64] into LDS with async B128 copies ----
  {
    // 256 16-byte chunks; each of 128 threads issues 2.
#pragma unroll
    for (int i = 0; i < 2; ++i) {
      const int c   = tid + i * 128;
      const int row = c >> 4;              // 0..15
      const int cc  = (c & 15) << 2;       // float offset within row, 16B chunks
      const float* gsrc = Qp + (size_t)(q0 + row) * DDM + hoff + cc;
      unsigned long long ga = (unsigned long long)(const void*)gsrc;
      unsigned lo = (unsigned)(unsigned long long)(const void*)(Qt + row * HDM + cc);
      asm volatile("global_load_async_to_lds_b128 %0, %1, off"
                   :: "v"(lo), "v"(ga) : "memory");
    }
    asm volatile("s_wait_asynccnt 0x0" ::: "memory");
  }
  __syncthreads();

  // ---- energy = (Q K^T)/scale * gamma^|i-j|, masked -> LDS ----
  const float* qrow = Qt + l * HDM + 2 * hf;     // A fragment from LDS
  for (int kt = w; kt < (SSQ >> 4); kt += 4) {
    const int k0 = kt << 4;
    const float* krow = Kp + (size_t)(k0 + l) * DDM + hoff + 2 * hf;
    v8f acc = {};
#pragma unroll
    for (int dd = 0; dd < HDM; dd += 4) {
      v2f a  = *(const v2f*)(qrow + dd);  // A[q][d..d+1] (LDS)
      v2f bf = *(const v2f*)(krow + dd);  // B[d][k] = K[k][d] (contiguous)
      acc = wmma_f32_4(a, bf, acc);
    }
    const int kcol = k0 + l;
    const int mv = mrow[kcol];
#pragma unroll
    for (int r = 0; r < 8; ++r) {
      const int row  = r + 8 * hf;
      const int qpos = q0 + row;
      float e = acc[r] * inv_scale;
      e *= exp2f(lg * fabsf((float)(qpos - kcol)));
      if (mv == 0) e = -1e10f;
      E[row * SSQ + kcol] = e;
    }
  }
  __syncthreads();

  // ---- row max & sum-exp: 8 lanes per row, shfl_xor combine (wave32) ----
  {
    const int row = tid >> 3;   // 0..15
    const int sub = tid & 7;    // 0..7, each covers 128 cols
    const float* er = E + row * SSQ + sub * 128;
    float pm = -3.4e38f;
    for (int c = 0; c < 128; ++c) pm = fmaxf(pm, er[c]);
#pragma unroll
    for (int off = 1; off < 8; off <<= 1) pm = fmaxf(pm, __shfl_xor(pm, off, 32));
    float ps = 0.0f;
    for (int c = 0; c < 128; ++c) ps += expf(er[c] - pm);
#pragma unroll
    for (int off = 1; off < 8; off <<= 1) ps += __shfl_xor(ps, off, 32);
    if (sub == 0) { red[row] = pm; red[16 + row] = 1.0f / ps; }
  }
  __syncthreads();

  // ---- normalize; store tile back to LDS + contiguous global write ----
  // stacked_attention[b][t][h][q0+row][col] : tile is one contiguous block
  float* abase = attn_out +
      ((((size_t)(b * TSC + t)) * HHN + h) * SSQ + q0) * SSQ;
  for (int idx = tid; idx < 16 * SSQ; idx += 128) {
    const int row = idx >> 10;
    const float v = expf(E[idx] - red[row]) * red[16 + row];
    E[idx] = v;
    abase[idx] = v;
  }
  __syncthreads();

  // ---- x = P @ V : wave w owns hd columns [16w, 16w+16) ----
  {
    const int d0 = w << 4;
    const float* eb   = E + (size_t)l * SSQ + 2 * hf;            // A from LDS
    const float* vcol = Vp + (size_t)(2 * hf) * DDM + hoff + d0 + l;
    v8f acc = {};
#pragma unroll 4
    for (int kk = 0; kk < SSQ; kk += 4) {
      v2f a = *(const v2f*)(eb + kk);
      v2f bf;
      bf.x = vcol[(size_t)kk * DDM];
      bf.y = vcol[(size_t)kk * DDM + DDM];
      acc = wmma_f32_4(a, bf, acc);
    }
#pragma unroll
    for (int r = 0; r < 8; ++r) {
      const int qpos = q0 + r + 8 * hf;
      xout[(size_t)(b * SSQ + qpos) * DH3 + t * DDM + hoff + d0 + l] = acc[r];
    }
  }
}

// ---------------------------------------------------------------------------
// In-place LayerNorm(1024) + ReLU over each row of H[2048,1024]
// ---------------------------------------------------------------------------
__global__ __launch_bounds__(256) void ln_relu_kernel(
    float* __restrict__ Hp, const float* __restrict__ lw,
    const float* __restrict__ lb) {
  __shared__ float rbuf[8];
  float* hrow = Hp + (size_t)blockIdx.x * DH2;
  const int tid = threadIdx.x;
  const int wv = tid >> 5, ln_ = tid & 31;

  float v[4];
  float s = 0.0f;
#pragma unroll
  for (int i = 0; i < 4; ++i) { v[i] = hrow[tid + i * 256]; s += v[i]; }
#pragma unroll
  for (int off = 16; off >= 1; off >>= 1) s += __shfl_xor(s, off, 32);
  if (ln_ == 0) rbuf[wv] = s;
  __syncthreads();
  float tot = 0.0f;
#pragma unroll
  for (int i = 0; i < 8; ++i) tot += rbuf[i];
  const float mu = tot * (1.0f / DH2);
  __syncthreads();

  float q = 0.0f;
#pragma unroll
  for (int i = 0; i < 4; ++i) { const float d = v[i] - mu; q += d * d; }
#pragma unroll
  for (int off = 16; off >= 1; off >>= 1) q += __shfl_xor(q, off, 32);
  if (ln_ == 0) rbuf[wv] = q;
  __syncthreads();
  float qt = 0.0f;
#pragma unroll
  for (int i = 0; i < 8; ++i) qt += rbuf[i];
  const float rstd = rsqrtf(qt * (1.0f / DH2) + 1e-5f);

#pragma unroll
  for (int i = 0; i < 4; ++i) {
    const int c = tid + i * 256;
    const float y = (v[i] - mu) * rstd * lw[c] + lb[c];
    hrow[c] = fmaxf(y, 0.0f);
  }
}

// ---------------------------------------------------------------------------
extern "C" void kernel_launch(void* const* d_in, const int* in_sizes, int n_in,
                              void* d_out, int out_size, void* d_ws,
                              size_t ws_size, hipStream_t stream) {
  (void)in_sizes; (void)n_in; (void)out_size; (void)ws_size;

  const float* query = (const float*)d_in[0];
  const float* key_  = (const float*)d_in[1];
  const float* value = (const float*)d_in[2];
  const int*   mask  = (const int*)d_in[3];
  const float* Wq = (const float*)d_in[4];
  const float* bq = (const float*)d_in[5];
  const float* Wk = (const float*)d_in[6];
  const float* bk = (const float*)d_in[7];
  const float* Wv = (const float*)d_in[8];
  const float* bv = (const float*)d_in[9];
  const float* gammas = (const float*)d_in[10];
  const float* W1 = (const float*)d_in[11];
  const float* b1 = (const float*)d_in[12];
  const float* lnw = (const float*)d_in[13];
  const float* lnb = (const float*)d_in[14];
  const float* W2 = (const float*)d_in[15];
  const float* b2 = (const float*)d_in[16];

  float* ws = (float*)d_ws;
  const size_t PROJ = (size_t)TSC * BB * SSQ * DDM;  // 3,145,728 floats
  float* Qb = ws;                 // [T,B,S,D]
  float* Kb = ws + PROJ;
  float* Vb = ws + 2 * PROJ;
  float* Xb = ws + 3 * PROJ;      // [B*S, 3D]
  float* Hb = Qb;                 // [B*S, 2D], aliases dead Q after attention

  float* outp  = (float*)d_out;                       // [B,S,D]
  float* attnp = outp + (size_t)BB * SSQ * DDM;       // [B,T,H,S,S]

  const int M = BB * SSQ;  // 2048

  // ---- Q/K/V projections: 9 GEMMs [2048,512] x [512,512] ----
  for (int t = 0; t < TSC; ++t) {
    const size_t wo = (size_t)t * DDM * DDM;
    const size_t bo = (size_t)t * DDM;
    const size_t po = (size_t)t * M * DDM;
    dim3 g((M / 16) * (DDM / 64));
    gemm16x64_f32<<<g, 32, 0, stream>>>(query, Wq + wo, bq + bo, Qb + po, M, DDM, DDM);
    gemm16x64_f32<<<g, 32, 0, stream>>>(key_,  Wk + wo, bk + bo, Kb + po, M, DDM, DDM);
    gemm16x64_f32<<<g, 32, 0, stream>>>(value, Wv + wo, bv + bo, Vb + po, M, DDM, DDM);
  }

  // ---- attention (energy + discount + mask + softmax + P@V) ----
  const size_t smem = (size_t)(16 * SSQ + 32 + 16 * HDM) * sizeof(float);
  attn_kernel<<<dim3(SSQ / 16, HHN, TSC * BB), 128, smem, stream>>>(
      Qb, Kb, Vb, mask, gammas, attnp, Xb);

  // ---- decoder: Linear(3D->2D) + LN + ReLU + Linear(2D->D) ----
  gemm16x64_f32<<<dim3((M / 16) * (DH2 / 64)), 32, 0, stream>>>(
      Xb, W1, b1, Hb, M, DH2, DH3);
  ln_relu_kernel<<<dim3(M), 256, 0, stream>>>(Hb, lnw, lnb);
  gemm16x64_f32<<<dim3((M / 16) * (DDM / 64)), 32, 0, stream>>>(
      Hb, W2, b2, outp, M, DDM, DH2);
}